// get_model_62414464745828
// MI455X (gfx1250) — compile-verified
//
#include <hip/hip_runtime.h>
#include <hip/hip_bf16.h>

// PointNet++ part-seg forward for MI455X (gfx1250, wave32).
// Row-major (B, Npts, C) feature tensors; GEMMs use V_WMMA_F32_16X16X4_F32.
// A buffers padded to Cs = round_up4(C): predication-free inner loop of
// 1x global_load_b64 (A, shared) + 2x ds b64 (W) + 2x wmma (two O-tiles).

typedef __attribute__((ext_vector_type(2))) float v2f;
typedef __attribute__((ext_vector_type(8))) float v8f;

#define NB 8          // batch
#define N0 8192       // input points
#define GEMM_WAVES 8  // 256 threads/block, 128 M-rows per block
#define MAXC 768      // largest (padded) input channel count (fp4 layer 0)

#if defined(__has_builtin)
#if __has_builtin(__builtin_amdgcn_global_load_async_to_lds_b32)
#define HAVE_ASYNC_LDS 1
#endif
#endif

typedef __attribute__((address_space(1))) int g_i32;
typedef __attribute__((address_space(3))) int l_i32;

static __device__ __forceinline__ void wait_async_lds() {
#if defined(HAVE_ASYNC_LDS)
#if __has_builtin(__builtin_amdgcn_s_wait_asynccnt)
    __builtin_amdgcn_s_wait_asynccnt(0);
#else
    asm volatile("s_wait_asynccnt 0x0" ::: "memory");
#endif
#endif
}

// ---------------------------------------------------------------------------
// (B,3,N) channel-major -> (B,N,3) row-major
__global__ void k_transpose_b3n(const float* __restrict__ in, float* __restrict__ out,
                                int B, int N) {
    int g = blockIdx.x * blockDim.x + threadIdx.x;
    if (g >= B * N) return;
    int b = g / N, n = g % N;
    const float* p = in + (size_t)b * 3 * N;
    float* o = out + (size_t)g * 3;
    o[0] = p[n]; o[1] = p[N + n]; o[2] = p[2 * N + n];
}

// ---------------------------------------------------------------------------
// Farthest point sampling: one block (256 thr) per batch, dist in registers.
#define FPS_T 256
__global__ void __launch_bounds__(FPS_T)
k_fps(const float* __restrict__ xyz, int* __restrict__ out, int N, int S) {
    int b = blockIdx.x;
    const float* P = xyz + (size_t)b * N * 3;
    int t = threadIdx.x;
    int per = (N + FPS_T - 1) / FPS_T;   // <= 32
    float dist[32];
    for (int j = 0; j < 32; ++j) dist[j] = 1e10f;

    __shared__ float sval[FPS_T];
    __shared__ int   sidx[FPS_T];
    __shared__ int   cur;
    if (t == 0) cur = 0;
    __syncthreads();

    for (int it = 0; it < S; ++it) {
        int far = cur;
        if (t == 0) out[b * S + it] = far;
        float cx = P[far * 3 + 0], cy = P[far * 3 + 1], cz = P[far * 3 + 2];
        float best = -1.0f; int bi = 0x7fffffff;
        for (int j = 0; j < per; ++j) {
            int i = t + j * FPS_T;
            if (i < N) {
                float dx = P[i * 3 + 0] - cx;
                float dy = P[i * 3 + 1] - cy;
                float dz = P[i * 3 + 2] - cz;
                float d  = dx * dx + dy * dy + dz * dz;
                float dm = fminf(dist[j], d);
                dist[j] = dm;
                if (dm > best) { best = dm; bi = i; }
            }
        }
        sval[t] = best; sidx[t] = bi;
        __syncthreads();
        for (int off = FPS_T / 2; off > 0; off >>= 1) {
            if (t < off) {
                if (sval[t + off] > sval[t] ||
                    (sval[t + off] == sval[t] && sidx[t + off] < sidx[t])) {
                    sval[t] = sval[t + off]; sidx[t] = sidx[t + off];
                }
            }
            __syncthreads();
        }
        if (t == 0) cur = sidx[0];
        __syncthreads();
    }
}

// ---------------------------------------------------------------------------
__global__ void k_gather_xyz(const float* __restrict__ xyz, const int* __restrict__ fidx,
                             float* __restrict__ out, int B, int N, int S) {
    int g = blockIdx.x * blockDim.x + threadIdx.x;
    if (g >= B * S) return;
    int b = g / S;
    int p = fidx[g];
    const float* s = xyz + ((size_t)b * N + p) * 3;
    float* o = out + (size_t)g * 3;
    o[0] = s[0]; o[1] = s[1]; o[2] = s[2];
}

// ---------------------------------------------------------------------------
// Ball query: keep the K lowest-index points with d^2 <= r^2, pad with first.
__global__ void k_ball_query(const float* __restrict__ xyz, const float* __restrict__ nxyz,
                             int* __restrict__ idx, int B, int N, int S, int K, float r2) {
    int g = blockIdx.x * blockDim.x + threadIdx.x;
    if (g >= B * S) return;
    int b = g / S;
    const float* P = xyz + (size_t)b * N * 3;
    float qx = nxyz[(size_t)g * 3 + 0];
    float qy = nxyz[(size_t)g * 3 + 1];
    float qz = nxyz[(size_t)g * 3 + 2];
    int* o = idx + (size_t)g * K;
    int cnt = 0;
    for (int i = 0; i < N && cnt < K; ++i) {
        float dx = P[i * 3 + 0] - qx;
        float dy = P[i * 3 + 1] - qy;
        float dz = P[i * 3 + 2] - qz;
        if (dx * dx + dy * dy + dz * dz <= r2) o[cnt++] = i;
    }
    int f = (cnt > 0) ? o[0] : 0;
    for (; cnt < K; ++cnt) o[cnt] = f;
}

// ---------------------------------------------------------------------------
// Build grouped rows: [xyz - center (3) | point features (Cp) | zero pad].
// Row stride Cs = round_up4(3+Cp).
__global__ void k_group(const float* __restrict__ xyz, const float* __restrict__ pts,
                        const float* __restrict__ nxyz, const int* __restrict__ idx,
                        float* __restrict__ out, int B, int N, int S, int K,
                        int Cp, int Cs) {
    int g = blockIdx.x * blockDim.x + threadIdx.x;
    if (g >= B * S * K) return;
    int bs = g / K;
    int b  = bs / S;
    int p  = idx[g];
    const float* src = xyz + ((size_t)b * N + p) * 3;
    const float* q   = nxyz + (size_t)bs * 3;
    float* o = out + (size_t)g * Cs;
    o[0] = src[0] - q[0]; o[1] = src[1] - q[1]; o[2] = src[2] - q[2];
    const float* f = pts + ((size_t)b * N + p) * Cp;
    for (int c = 0; c < Cp; ++c) o[3 + c] = f[c];
    for (int c = 3 + Cp; c < Cs; ++c) o[c] = 0.0f;
}

// ---------------------------------------------------------------------------
// Y[M,O] = A[M,Cs] * W[O,C]^T + bias  via V_WMMA_F32_16X16X4_F32.
// Each wave computes TWO 16x16 output tiles (columns o0..o0+15 and
// o0+16..o0+31) sharing one A fragment per K-step: the inner loop is
// 1x global_load_b64 (A) + 2x LDS b64 (W) + 2x wmma -> 16 flops per byte
// of global traffic (vs 8 with a single tile).
// A lane layout (16x4 fp32): lanes 0-15 row M=lane (K=c0,c0+1),
// lanes 16-31 row M=lane-16 (K=c0+2,c0+3).
// B (4x16): lane&15 = column; lanes 0-15 carry K=c0,c0+1, lanes 16-31
// carry K=c0+2,c0+3 -> contiguous pair per lane.
// C/D: VGPR g -> row m0+g (lanes 0-15) / m0+8+g (lanes 16-31), col lane&15.
__global__ void __launch_bounds__(GEMM_WAVES * 32)
k_gemm_bias_wmma(const float* __restrict__ A, const float* __restrict__ W,
                 const float* __restrict__ bias, float* __restrict__ Y,
                 int M, int C, int Cs, int O) {
    __shared__ float Wt[32 * MAXC];          // 32 output channels staged
    int o0 = blockIdx.y * 32;
    for (int r = 0; r < 32; ++r) {
        const float* wrow = W + (size_t)(o0 + r) * C;
        for (int c = threadIdx.x; c < Cs; c += blockDim.x) {
            if (c < C) {
#if defined(HAVE_ASYNC_LDS)
                __builtin_amdgcn_global_load_async_to_lds_b32(
                    (g_i32*)const_cast<float*>(wrow + c),
                    (l_i32*)&Wt[r * Cs + c],
                    0, 0);
#else
                Wt[r * Cs + c] = wrow[c];
#endif
            } else {
                Wt[r * Cs + c] = 0.0f;
            }
        }
    }
    wait_async_lds();
    __syncthreads();

    int wave = threadIdx.x >> 5;
    int lane = threadIdx.x & 31;
    int hf2  = (lane >> 4) << 1;   // 0 or 2
    int l    = lane & 15;
    int m0   = (blockIdx.x * GEMM_WAVES + wave) * 16;
    int arow = m0 + l;
    int arowc = arow < M ? arow : M - 1;         // clamp, stores are guarded
    const float* Ab = A + (size_t)arowc * Cs;

    __builtin_prefetch(Ab, 0, 1);
    __builtin_prefetch(Ab + 64, 0, 1);

    v8f acc0, acc1;
    float bi0 = bias[o0 + l];
    float bi1 = bias[o0 + 16 + l];
    for (int i = 0; i < 8; ++i) { acc0[i] = bi0; acc1[i] = bi1; }

    const float* Wrow0 = &Wt[(size_t)l * Cs];
    const float* Wrow1 = &Wt[(size_t)(l + 16) * Cs];
    for (int c0 = 0; c0 < Cs; c0 += 4) {
        v2f a  = *(const v2f*)(Ab + c0 + hf2);
        v2f b0 = *(const v2f*)(Wrow0 + c0 + hf2);
        v2f b1 = *(const v2f*)(Wrow1 + c0 + hf2);
        acc0 = __builtin_amdgcn_wmma_f32_16x16x4_f32(
            false, a, false, b0, (short)0, acc0, false, false);
        acc1 = __builtin_amdgcn_wmma_f32_16x16x4_f32(
            false, a, false, b1, (short)0, acc1, false, false);
    }

    int oc0 = o0 + l;
    int oc1 = o0 + 16 + l;
    int mb  = m0 + ((lane >> 4) << 3);
    if (m0 + 16 <= M) {                           // wave-uniform fast path
        for (int r = 0; r < 8; ++r) {
            size_t row = (size_t)(mb + r) * O;
            Y[row + oc0] = acc0[r];
            Y[row + oc1] = acc1[r];
        }
    } else {
        for (int r = 0; r < 8; ++r) {
            int mr = mb + r;
            if (mr < M) {
                size_t row = (size_t)mr * O;
                Y[row + oc0] = acc0[r];
                Y[row + oc1] = acc1[r];
            }
        }
    }
}

// ---------------------------------------------------------------------------
// Per-channel batch stats over all M rows -> scale/shift. One block / channel.
__global__ void __launch_bounds__(256)
k_bn_stats(const float* __restrict__ Y, const float* __restrict__ g,
           const float* __restrict__ be, float* __restrict__ scale,
           float* __restrict__ shift, int M, int O) {
    int o = blockIdx.x;
    float s = 0.0f, ss = 0.0f;
    for (int i = threadIdx.x; i < M; i += blockDim.x) {
        float v = Y[(size_t)i * O + o];
        s += v; ss += v * v;
    }
    __shared__ float s1[256], s2[256];
    s1[threadIdx.x] = s; s2[threadIdx.x] = ss;
    __syncthreads();
    for (int off = 128; off > 0; off >>= 1) {
        if (threadIdx.x < off) {
            s1[threadIdx.x] += s1[threadIdx.x + off];
            s2[threadIdx.x] += s2[threadIdx.x + off];
        }
        __syncthreads();
    }
    if (threadIdx.x == 0) {
        float mean = s1[0] / (float)M;
        float var  = s2[0] / (float)M - mean * mean;
        float sc   = g[o] * rsqrtf(var + 1e-5f);
        scale[o] = sc;
        shift[o] = be[o] - mean * sc;
    }
}

__global__ void k_bn_apply(float* __restrict__ Y, const float* __restrict__ scale,
                           const float* __restrict__ shift, long long total, int O,
                           int relu) {
    long long g = (long long)blockIdx.x * blockDim.x + threadIdx.x;
    if (g >= total) return;
    int o = (int)(g % O);
    float v = Y[g] * scale[o] + shift[o];
    if (relu) v = fmaxf(v, 0.0f);
    Y[g] = v;
}

// ---------------------------------------------------------------------------
__global__ void k_maxpool(const float* __restrict__ Y, float* __restrict__ out,
                          int BS, int K, int O) {
    int g = blockIdx.x * blockDim.x + threadIdx.x;
    if (g >= BS * O) return;
    int o = g % O, bs = g / O;
    const float* p = Y + (size_t)bs * K * O + o;
    float m = p[0];
    for (int k = 1; k < K; ++k) m = fmaxf(m, p[(size_t)k * O]);
    out[g] = m;
}

// ---------------------------------------------------------------------------
// 3-NN with inverse-distance weights (matches lax.top_k(-d,3) ordering).
__global__ void k_three_nn(const float* __restrict__ xyz1, const float* __restrict__ xyz2,
                           int* __restrict__ nidx, float* __restrict__ nw,
                           int B, int S1, int S2) {
    int g = blockIdx.x * blockDim.x + threadIdx.x;
    if (g >= B * S1) return;
    int b = g / S1;
    const float* q = xyz1 + (size_t)g * 3;
    const float* P = xyz2 + (size_t)b * S2 * 3;
    float qx = q[0], qy = q[1], qz = q[2];
    float d0 = 1e30f, d1 = 1e30f, d2 = 1e30f;
    int i0 = 0, i1 = 0, i2 = 0;
    for (int j = 0; j < S2; ++j) {
        float dx = P[j * 3 + 0] - qx;
        float dy = P[j * 3 + 1] - qy;
        float dz = P[j * 3 + 2] - qz;
        float d = dx * dx + dy * dy + dz * dz;
        if (d < d0)      { d2 = d1; i2 = i1; d1 = d0; i1 = i0; d0 = d; i0 = j; }
        else if (d < d1) { d2 = d1; i2 = i1; d1 = d; i1 = j; }
        else if (d < d2) { d2 = d; i2 = j; }
    }
    float r0 = 1.0f / (d0 + 1e-8f);
    float r1 = 1.0f / (d1 + 1e-8f);
    float r2 = 1.0f / (d2 + 1e-8f);
    float inv = 1.0f / (r0 + r1 + r2);
    nidx[(size_t)g * 3 + 0] = i0; nw[(size_t)g * 3 + 0] = r0 * inv;
    nidx[(size_t)g * 3 + 1] = i1; nw[(size_t)g * 3 + 1] = r1 * inv;
    nidx[(size_t)g * 3 + 2] = i2; nw[(size_t)g * 3 + 2] = r2 * inv;
}

// Build FP input rows: [pts1 (C1) | interpolated pts2 (C2)]  (C1+C2 % 4 == 0)
__global__ void k_interp_concat(const float* __restrict__ pts1, const float* __restrict__ pts2,
                                const int* __restrict__ nidx, const float* __restrict__ nw,
                                float* __restrict__ x, int B, int S1, int S2,
                                int C1, int C2) {
    long long t = (long long)blockIdx.x * blockDim.x + threadIdx.x;
    int Cout = C1 + C2;
    long long total = (long long)B * S1 * Cout;
    if (t >= total) return;
    int c = (int)(t % Cout);
    long long g = t / Cout;
    int b = (int)(g / S1);
    if (c < C1) {
        x[t] = pts1[g * C1 + c];
    } else {
        int cc = c - C1;
        const float* base = pts2 + (size_t)b * S2 * C2;
        const int*   ii = nidx + g * 3;
        const float* ww = nw + g * 3;
        x[t] = ww[0] * base[(size_t)ii[0] * C2 + cc]
             + ww[1] * base[(size_t)ii[1] * C2 + cc]
             + ww[2] * base[(size_t)ii[2] * C2 + cc];
    }
}

// ---------------------------------------------------------------------------
__global__ void k_conv2_sigmoid(const float* __restrict__ X, const float* __restrict__ W,
                                const float* __restrict__ bias, float* __restrict__ out,
                                int M, int C) {
    int g = blockIdx.x * blockDim.x + threadIdx.x;
    if (g >= M) return;
    const float* x = X + (size_t)g * C;
    float acc = bias[0];
    for (int c = 0; c < C; ++c) acc += x[c] * W[c];
    out[g] = 1.0f / (1.0f + expf(-acc));
}

// ===========================================================================
// Host orchestration
// ===========================================================================
struct LW { const float *W, *b, *g, *be; int C, O; };

static inline int pad4(int c) { return (c + 3) & ~3; }

static void run_mlp_layer(hipStream_t st, const float* X, const LW& lw, float* Y,
                          int M, bool relu, float* scale, float* shift) {
    int Cs = pad4(lw.C);   // producers guarantee the padded stride
    dim3 grid((M + GEMM_WAVES * 16 - 1) / (GEMM_WAVES * 16), lw.O / 32);
    k_gemm_bias_wmma<<<grid, GEMM_WAVES * 32, 0, st>>>(X, lw.W, lw.b, Y, M, lw.C, Cs, lw.O);
    k_bn_stats<<<lw.O, 256, 0, st>>>(Y, lw.g, lw.be, scale, shift, M, lw.O);
    long long tot = (long long)M * lw.O;
    k_bn_apply<<<(int)((tot + 255) / 256), 256, 0, st>>>(Y, scale, shift, tot, lw.O, relu ? 1 : 0);
}

static void run_sa(hipStream_t st, const float* xyz_rm, const float* pts, int Nl, int Cp,
                   int S, int K, float radius, const LW* lws, int nl,
                   float* new_xyz, float* pooled, int* fpsIdx, int* ballIdx,
                   float* sA, float* sB, float* sC, float* scale, float* shift) {
    k_fps<<<NB, FPS_T, 0, st>>>(xyz_rm, fpsIdx, Nl, S);
    int bs = NB * S;
    k_gather_xyz<<<(bs + 255) / 256, 256, 0, st>>>(xyz_rm, fpsIdx, new_xyz, NB, Nl, S);
    k_ball_query<<<(bs + 255) / 256, 256, 0, st>>>(xyz_rm, new_xyz, ballIdx,
                                                   NB, Nl, S, K, radius * radius);
    int bsk = bs * K;
    int Cs0 = pad4(3 + Cp);
    k_group<<<(bsk + 255) / 256, 256, 0, st>>>(xyz_rm, pts, new_xyz, ballIdx, sA,
                                               NB, Nl, S, K, Cp, Cs0);
    int M = bsk;
    const float* x = sA;
    float* y = sB;
    for (int i = 0; i < nl; ++i) {
        run_mlp_layer(st, x, lws[i], y, M, true, scale, shift);
        x = y;
        y = (y == sB) ? sC : sB;
    }
    int tot = bs * lws[nl - 1].O;
    k_maxpool<<<(tot + 255) / 256, 256, 0, st>>>(x, pooled, bs, K, lws[nl - 1].O);
}

static void run_fp(hipStream_t st, const float* xyz1, int S1, const float* xyz2, int S2,
                   const float* pts1, int C1, const float* pts2, int C2,
                   const LW* lws, int nl, float* outbuf,
                   int* nnIdx, float* nnW, float* sA, float* sB, float* sC,
                   float* scale, float* shift) {
    int bs = NB * S1;
    k_three_nn<<<(bs + 255) / 256, 256, 0, st>>>(xyz1, xyz2, nnIdx, nnW, NB, S1, S2);
    int Cout = C1 + C2;   // all FP concat widths are multiples of 4
    long long tot = (long long)bs * Cout;
    k_interp_concat<<<(int)((tot + 255) / 256), 256, 0, st>>>(pts1, pts2, nnIdx, nnW, sA,
                                                              NB, S1, S2, C1, C2);
    int M = bs;
    const float* x = sA;
    float* y = sB;
    for (int i = 0; i < nl; ++i) {
        float* dst = (i == nl - 1) ? outbuf : y;
        run_mlp_layer(st, x, lws[i], dst, M, true, scale, shift);
        x = dst;
        if (dst == y) y = (y == sB) ? sC : sB;
    }
}

extern "C" void kernel_launch(void* const* d_in, const int* in_sizes, int n_in,
                              void* d_out, int out_size, void* d_ws, size_t ws_size,
                              hipStream_t stream) {
    (void)in_sizes; (void)n_in; (void)out_size; (void)ws_size;
    const float* xyz = (const float*)d_in[0];

    // Params flattened in setup_inputs() insertion order; each layer: W,b,g,be.
    int p = 1;
    auto take = [&](int C, int O) {
        LW l{(const float*)d_in[p], (const float*)d_in[p + 1],
             (const float*)d_in[p + 2], (const float*)d_in[p + 3], C, O};
        p += 4;
        return l;
    };
    LW sa1[2] = { take(6, 32),    take(32, 64)   };
    LW sa2[2] = { take(67, 64),   take(64, 128)  };
    LW sa3[2] = { take(131, 128), take(128, 256) };
    LW sa4[2] = { take(259, 256), take(256, 512) };
    LW fp4[2] = { take(768, 256), take(256, 256) };
    LW fp3[2] = { take(384, 256), take(256, 256) };
    LW fp2[2] = { take(320, 256), take(256, 128) };
    LW fp1[3] = { take(128, 128), take(128, 128), take(128, 128) };
    const float* conv1_W = (const float*)d_in[p++];
    const float* conv1_b = (const float*)d_in[p++];
    const float* bn1_g   = (const float*)d_in[p++];
    const float* bn1_be  = (const float*)d_in[p++];
    const float* conv2_W = (const float*)d_in[p++];
    const float* conv2_b = (const float*)d_in[p++];

    // ---- workspace carve-up (floats) ----
    float* wsf = (float*)d_ws;
    size_t off = 0;
    auto alloc = [&](size_t n) { float* q = wsf + off; off += n; return q; };
    float* xyz_rm = alloc((size_t)NB * N0 * 3);
    float* nx1 = alloc((size_t)NB * 2048 * 3);
    float* nx2 = alloc((size_t)NB * 2048 * 3);
    float* nx3 = alloc((size_t)NB * 1024 * 3);
    float* nx4 = alloc((size_t)NB * 1024 * 3);
    float* l1p = alloc((size_t)NB * 2048 * 64);
    float* l2p = alloc((size_t)NB * 2048 * 128);
    float* l3p = alloc((size_t)NB * 1024 * 256);
    float* l4p = alloc((size_t)NB * 1024 * 512);
    float* f3  = alloc((size_t)NB * 1024 * 256);   // fp4 out
    float* f2  = alloc((size_t)NB * 2048 * 256);   // fp3 out
    float* f1  = alloc((size_t)NB * 2048 * 128);   // fp2 out
    float* f0  = alloc((size_t)NB * 8192 * 128);   // fp1 out
    float* c1  = alloc((size_t)NB * 8192 * 128);   // conv1 out
    float* scale = alloc(1024);
    float* shift = alloc(1024);
    float* nnW   = alloc((size_t)NB * 8192 * 3);
    int* fpsIdx  = (int*)alloc((size_t)NB * 2048);
    int* ballIdx = (int*)alloc((size_t)262144);
    int* nnIdx   = (int*)alloc((size_t)NB * 8192 * 3);
    float* sA = alloc((size_t)68157440);   // grouped (padded) / fp concat input
    float* sB = alloc((size_t)67108864);   // mlp ping
    float* sC = alloc((size_t)134217728);  // mlp pong

    // ---- pipeline ----
    k_transpose_b3n<<<(NB * N0 + 255) / 256, 256, 0, stream>>>(xyz, xyz_rm, NB, N0);

    run_sa(stream, xyz_rm, xyz_rm, N0,   3,   2048,  8, 0.2f, sa1, 2, nx1, l1p,
           fpsIdx, ballIdx, sA, sB, sC, scale, shift);
    run_sa(stream, nx1,    l1p,    2048, 64,  2048, 16, 0.4f, sa2, 2, nx2, l2p,
           fpsIdx, ballIdx, sA, sB, sC, scale, shift);
    run_sa(stream, nx2,    l2p,    2048, 128, 1024, 32, 0.8f, sa3, 2, nx3, l3p,
           fpsIdx, ballIdx, sA, sB, sC, scale, shift);
    run_sa(stream, nx3,    l3p,    1024, 256, 1024, 32, 1.0f, sa4, 2, nx4, l4p,
           fpsIdx, ballIdx, sA, sB, sC, scale, shift);

    run_fp(stream, nx3, 1024, nx4, 1024, l3p, 256, l4p, 512, fp4, 2, f3,
           nnIdx, nnW, sA, sB, sC, scale, shift);
    run_fp(stream, nx2, 2048, nx3, 1024, l2p, 128, f3,  256, fp3, 2, f2,
           nnIdx, nnW, sA, sB, sC, scale, shift);
    run_fp(stream, nx1, 2048, nx2, 2048, l1p, 64,  f2,  256, fp2, 2, f1,
           nnIdx, nnW, sA, sB, sC, scale, shift);
    run_fp(stream, xyz_rm, 8192, nx1, 2048, nullptr, 0, f1, 128, fp1, 3, f0,
           nnIdx, nnW, sA, sB, sC, scale, shift);

    // conv1 (128->128) + BN (no relu)
    LW c1l{conv1_W, conv1_b, bn1_g, bn1_be, 128, 128};
    run_mlp_layer(stream, f0, c1l, c1, NB * 8192, /*relu=*/false, scale, shift);

    // conv2 (128->1) + sigmoid -> (B, N, 1) flat
    k_conv2_sigmoid<<<(NB * 8192 + 255) / 256, 256, 0, stream>>>(
        c1, conv2_W, conv2_b, (float*)d_out, NB * 8192, 128);
}